// MyGATConv_35536559407735
// MI455X (gfx1250) — compile-verified
//
#include <hip/hip_runtime.h>
#include <math.h>

#define SLOPE   0.2f
#define IN_CH   256
#define HEADS   4
#define OUT_CH  64
#define HC      256   // HEADS*OUT_CH

typedef __attribute__((ext_vector_type(2))) float v2f;
typedef __attribute__((ext_vector_type(8))) float v8f;

// ---------------------------------------------------------------------------
// Kernel 1: XS[N, 256] = X[N, 256] @ W[256, 256]^T using V_WMMA_F32_16X16X4_F32
// Each wave computes a 64-row x 16-col panel (4 row-tiles, 4 accumulators),
// reusing each B fragment across 4 WMMAs (5 b64 loads : 4 WMMAs).
// A 16x4 f32 layout: lanes 0-15 -> K=0/1 in VGPR0/1; lanes 16-31 -> K=2/3.
// C/D layout: VGPR v -> row = v + 8*(lane>>4), col = lane&15.
// ---------------------------------------------------------------------------
__global__ __launch_bounds__(256) void gat_proj_wmma(const float* __restrict__ X,
                                                     const float* __restrict__ W,
                                                     float* __restrict__ XS,
                                                     int nRows, int nGroups) {
  const int lane = threadIdx.x & 31;
  const int wave = threadIdx.x >> 5;
  const int grp  = blockIdx.x * 8 + wave;
  if (grp >= nGroups) return;                 // wave-uniform early-out (EXEC stays full)

  const int rowBase = (grp >> 4) * 64;        // 4 row tiles of 16
  const int colTile = grp & 15;               // 256/16 = 16 col tiles
  const int half = lane >> 4;                 // 0: K=0,1   1: K=2,3
  const int l16  = lane & 15;

  const int bcol = colTile * 16 + l16;        // column of XS == row of W
  const float* __restrict__ wrow = W + (size_t)bcol * IN_CH;

  const float* xr[4];
#pragma unroll
  for (int i = 0; i < 4; ++i) {
    int r = rowBase + i * 16 + l16;
    if (r >= nRows) r = nRows - 1;            // clamp loads; stores handled below
    xr[i] = X + (size_t)r * IN_CH;
  }

  v8f acc[4];
#pragma unroll
  for (int i = 0; i < 4; ++i) acc[i] = (v8f){0.f,0.f,0.f,0.f,0.f,0.f,0.f,0.f};

#pragma unroll 4
  for (int k = 0; k < IN_CH; k += 4) {
    const int ka = k + 2 * half;
    const v2f b  = *(const v2f*)(wrow + ka);   // shared across 4 row tiles
    const v2f a0 = *(const v2f*)(xr[0] + ka);
    const v2f a1 = *(const v2f*)(xr[1] + ka);
    const v2f a2 = *(const v2f*)(xr[2] + ka);
    const v2f a3 = *(const v2f*)(xr[3] + ka);
    acc[0] = __builtin_amdgcn_wmma_f32_16x16x4_f32(false, a0, false, b, (short)0, acc[0], false, false);
    acc[1] = __builtin_amdgcn_wmma_f32_16x16x4_f32(false, a1, false, b, (short)0, acc[1], false, false);
    acc[2] = __builtin_amdgcn_wmma_f32_16x16x4_f32(false, a2, false, b, (short)0, acc[2], false, false);
    acc[3] = __builtin_amdgcn_wmma_f32_16x16x4_f32(false, a3, false, b, (short)0, acc[3], false, false);
  }

  const int col = colTile * 16 + l16;
  if (rowBase + 64 <= nRows) {
    // Common path: whole 64-row panel in range, unpredicated stores.
#pragma unroll
    for (int i = 0; i < 4; ++i) {
      const size_t rbase = (size_t)(rowBase + i * 16 + 8 * half);
#pragma unroll
      for (int v = 0; v < 8; ++v)
        XS[(rbase + v) * HC + col] = acc[i][v];
    }
  } else {
    // Tail panel only: per-row guard.
#pragma unroll
    for (int i = 0; i < 4; ++i) {
#pragma unroll
      for (int v = 0; v < 8; ++v) {
        const int r = rowBase + i * 16 + v + 8 * half;
        if (r < nRows) XS[(size_t)r * HC + col] = acc[i][v];
      }
    }
  }
}

// ---------------------------------------------------------------------------
// Kernel 2: per-(node, head) attention logits
//   a_src[n,h] = sum_c xs[n,h,c] * att_src[h,c]
//   a_dst[n,h] = sum_c xs[n,h,c] * att_dst[h,c]
// ---------------------------------------------------------------------------
__global__ __launch_bounds__(256) void gat_logits(const float* __restrict__ XS,
                                                  const float* __restrict__ att_s,
                                                  const float* __restrict__ att_d,
                                                  float* __restrict__ a_src,
                                                  float* __restrict__ a_dst,
                                                  int nh) {
  const int t = blockIdx.x * 256 + threadIdx.x;
  if (t >= nh) return;
  const int h = t & (HEADS - 1);
  const float* __restrict__ row = XS + (size_t)(t >> 2) * HC + h * OUT_CH;
  const float* __restrict__ ws  = att_s + h * OUT_CH;
  const float* __restrict__ wd  = att_d + h * OUT_CH;
  float ss = 0.f, sd = 0.f;
#pragma unroll
  for (int cc = 0; cc < OUT_CH; cc += 4) {
    float4 xv = *(const float4*)(row + cc);
    float4 sv = *(const float4*)(ws + cc);
    float4 dv = *(const float4*)(wd + cc);
    ss += xv.x * sv.x + xv.y * sv.y + xv.z * sv.z + xv.w * sv.w;
    sd += xv.x * dv.x + xv.y * dv.y + xv.z * dv.z + xv.w * dv.w;
  }
  a_src[t] = ss;
  a_dst[t] = sd;
}

// ---------------------------------------------------------------------------
// Kernel 3: per-dst CSR segment softmax + weighted aggregation + head mean.
// One workgroup (256 thr = 8 wave32) per dst node; thread t owns (h,c).
// XS (51 MB) is L2-resident on MI455X (192 MB L2) -> the 820 MB edge gather
// is an L2-bandwidth problem, organized as 128B-coalesced reads per wave.
// ---------------------------------------------------------------------------
__global__ __launch_bounds__(256) void gat_aggregate(const float* __restrict__ XS,
                                                     const float* __restrict__ a_src,
                                                     const float* __restrict__ a_dst,
                                                     const int* __restrict__ ptr,
                                                     const int* __restrict__ idx,
                                                     const float* __restrict__ bias,
                                                     float* __restrict__ out) {
  const int d = blockIdx.x;
  const int t = threadIdx.x;
  const int h = t >> 6;
  const int c = t & 63;
  const int e0 = ptr[d];
  const int e1 = ptr[d + 1];

  __shared__ float4 red[256];
  __shared__ float  s_m[HEADS];
  __shared__ float  s_inv[HEADS];
  __shared__ float  s_ad[HEADS];
  __shared__ float  s_w[64 * HEADS];
  __shared__ int    s_src[64];
  __shared__ float  fin[256];

  if (t < HEADS) s_ad[t] = a_dst[(size_t)d * HEADS + t];
  __syncthreads();
  const float ad0 = s_ad[0], ad1 = s_ad[1], ad2 = s_ad[2], ad3 = s_ad[3];

  // ---- Pass A: segment max of leaky-relu logits -----------------------------
  float4 mx = make_float4(-INFINITY, -INFINITY, -INFINITY, -INFINITY);
  for (int e = e0 + t; e < e1; e += 256) {
    const int s = idx[e];
    float4 as = *(const float4*)(a_src + (size_t)s * HEADS);
    float a0 = as.x + ad0; a0 = (a0 >= 0.f) ? a0 : SLOPE * a0;
    float a1 = as.y + ad1; a1 = (a1 >= 0.f) ? a1 : SLOPE * a1;
    float a2 = as.z + ad2; a2 = (a2 >= 0.f) ? a2 : SLOPE * a2;
    float a3 = as.w + ad3; a3 = (a3 >= 0.f) ? a3 : SLOPE * a3;
    mx.x = fmaxf(mx.x, a0); mx.y = fmaxf(mx.y, a1);
    mx.z = fmaxf(mx.z, a2); mx.w = fmaxf(mx.w, a3);
  }
  red[t] = mx;
  __syncthreads();
  for (int off = 128; off > 0; off >>= 1) {
    if (t < off) {
      float4 m = red[t], o = red[t + off];
      m.x = fmaxf(m.x, o.x); m.y = fmaxf(m.y, o.y);
      m.z = fmaxf(m.z, o.z); m.w = fmaxf(m.w, o.w);
      red[t] = m;
    }
    __syncthreads();
  }
  if (t == 0) {
    float4 m = red[0];
    s_m[0] = m.x; s_m[1] = m.y; s_m[2] = m.z; s_m[3] = m.w;
  }
  __syncthreads();
  const float m0 = s_m[0], m1 = s_m[1], m2 = s_m[2], m3 = s_m[3];

  // ---- Pass B: sum of exp ---------------------------------------------------
  float4 sm = make_float4(0.f, 0.f, 0.f, 0.f);
  for (int e = e0 + t; e < e1; e += 256) {
    const int s = idx[e];
    float4 as = *(const float4*)(a_src + (size_t)s * HEADS);
    float a0 = as.x + ad0; a0 = (a0 >= 0.f) ? a0 : SLOPE * a0;
    float a1 = as.y + ad1; a1 = (a1 >= 0.f) ? a1 : SLOPE * a1;
    float a2 = as.z + ad2; a2 = (a2 >= 0.f) ? a2 : SLOPE * a2;
    float a3 = as.w + ad3; a3 = (a3 >= 0.f) ? a3 : SLOPE * a3;
    sm.x += __expf(a0 - m0); sm.y += __expf(a1 - m1);
    sm.z += __expf(a2 - m2); sm.w += __expf(a3 - m3);
  }
  red[t] = sm;
  __syncthreads();
  for (int off = 128; off > 0; off >>= 1) {
    if (t < off) {
      float4 m = red[t], o = red[t + off];
      m.x += o.x; m.y += o.y; m.z += o.z; m.w += o.w;
      red[t] = m;
    }
    __syncthreads();
  }
  if (t == 0) {
    float4 s = red[0];
    s_inv[0] = 1.f / (s.x + 1e-16f); s_inv[1] = 1.f / (s.y + 1e-16f);
    s_inv[2] = 1.f / (s.z + 1e-16f); s_inv[3] = 1.f / (s.w + 1e-16f);
  }
  __syncthreads();

  // ---- Pass C: chunked weighted aggregation (64 edges staged in LDS) --------
  float acc = 0.f;
  for (int base = e0; base < e1; base += 64) {
    const int n = min(64, e1 - base);
    if (t < n * HEADS) {
      const int el = t >> 2;
      const int hh = t & 3;
      const int s  = idx[base + el];
      if (hh == 0) s_src[el] = s;
      float a = a_src[(size_t)s * HEADS + hh] + s_ad[hh];
      a = (a >= 0.f) ? a : SLOPE * a;
      s_w[el * HEADS + hh] = __expf(a - s_m[hh]) * s_inv[hh];
      __builtin_prefetch(XS + (size_t)s * HC + hh * OUT_CH, 0, 0);  // global_prefetch_b8
    }
    __syncthreads();
    for (int j = 0; j < n; ++j) {
      acc += s_w[j * HEADS + h] * XS[(size_t)s_src[j] * HC + h * OUT_CH + c];
    }
    __syncthreads();
  }

  // ---- head mean + bias -----------------------------------------------------
  fin[t] = acc;
  __syncthreads();
  if (t < OUT_CH) {
    float v = (fin[t] + fin[64 + t] + fin[128 + t] + fin[192 + t]) * 0.25f + bias[t];
    out[(size_t)d * OUT_CH + t] = v;
  }
}

// ---------------------------------------------------------------------------
extern "C" void kernel_launch(void* const* d_in, const int* in_sizes, int n_in,
                              void* d_out, int out_size, void* d_ws, size_t ws_size,
                              hipStream_t stream) {
  const float* x     = (const float*)d_in[0];
  const float* lin   = (const float*)d_in[1];
  const float* att_s = (const float*)d_in[2];
  const float* att_d = (const float*)d_in[3];
  const float* bias  = (const float*)d_in[4];
  const int*   ptr   = (const int*)d_in[5];
  const int*   idx   = (const int*)d_in[6];

  const int N = in_sizes[0] / IN_CH;     // num nodes
  // workspace layout: xs[N*256] f32 | a_src[N*4] | a_dst[N*4]
  float* xs    = (float*)d_ws;
  float* a_src = xs + (size_t)N * HC;
  float* a_dst = a_src + (size_t)N * HEADS;

  const int rowGroups = (N + 63) / 64;           // 64 rows per wave
  const int nGroups   = rowGroups * (HC / 16);   // x 16 col tiles
  const int blocks1   = (nGroups + 7) / 8;       // 8 waves per block
  gat_proj_wmma<<<blocks1, 256, 0, stream>>>(x, lin, xs, N, nGroups);

  const int nh = N * HEADS;
  gat_logits<<<(nh + 255) / 256, 256, 0, stream>>>(xs, att_s, att_d, a_src, a_dst, nh);

  gat_aggregate<<<N, 256, 0, stream>>>(xs, a_src, a_dst, ptr, idx, bias, (float*)d_out);

  (void)n_in; (void)out_size; (void)ws_size;
}